// CNODEmod_49143015801408
// MI455X (gfx1250) — compile-verified
//
#include <hip/hip_runtime.h>
#include <hip/hip_bf16.h>
#include <math.h>

typedef __attribute__((ext_vector_type(2))) float v2f;
typedef __attribute__((ext_vector_type(8))) float v8f;

#define NCH    64
#define HIDN   64
#define TSTEPS 8
#define BATCHN 1024
#define DTC    0.02f
#define DELTAC 5.0f
#define NPRE   40
#define NPOST  10
#define NUPD   10
#define LDSW   68   // padded LDS row stride in floats (bank-conflict free)

// D = A(16x4 f32) * B(4x16 f32) + C  on the matrix pipe
__device__ __forceinline__ v8f wmma4(v2f a, v2f b, v8f c) {
  return __builtin_amdgcn_wmma_f32_16x16x4_f32(false, a, false, b, (short)0, c,
                                               false, false);
}

// sum across the 16 lanes of this half-wave (lanes 0-15 or 16-31)
__device__ __forceinline__ float halfred(float p) {
  p += __shfl_xor(p, 1, 32);
  p += __shfl_xor(p, 2, 32);
  p += __shfl_xor(p, 4, 32);
  p += __shfl_xor(p, 8, 32);
  return p;
}

// Block = 4 waves = 2 wave-pairs; each pair owns 16 batch rows.
//   w0 (even wave): evolve GEMM H@E, rank-1 combine, update loop, merge
//   w1 (odd wave):  MLP GEMM H@W1 -> relu -> .W2, publishes y via LDS; unc head
__global__ void __launch_bounds__(128)
cnode_kernel(const float* __restrict__ times, const float* __restrict__ Y,
             const float* __restrict__ mask,
             const float* __restrict__ W1, const float* __restrict__ b1,
             const float* __restrict__ W2, const float* __restrict__ b2,
             const float* __restrict__ U1, const float* __restrict__ ub1,
             const float* __restrict__ U2, const float* __restrict__ ub2,
             float* __restrict__ out)
{
  extern __shared__ float lds[];
  float* Et  = lds;                 // [64][LDSW]  Et[n][k] = I + DT*A^T (B-operand layout)
  float* W1t = lds + 64 * LDSW;     // [64][LDSW]  W1t[h][k] = W1[k][h]
  float* U1t = lds + 2 * 64 * LDSW; // [64][LDSW]  U1t[h][k] = U1[k][h]

  const int tid   = threadIdx.x;
  const int lane  = tid & 31;
  const int wave  = tid >> 5;
  const int pair  = wave >> 1;
  const bool isW0 = (wave & 1) == 0;
  const int lmod  = lane & 15;
  const int lhalf = lane >> 4;

  float* Hb   = lds + 3 * 64 * LDSW + pair * (16 * LDSW);                   // state
  float* Hp   = lds + 3 * 64 * LDSW + 2 * (16 * LDSW) + pair * (16 * LDSW); // h_pre
  float* ybuf = lds + 3 * 64 * LDSW + 4 * (16 * LDSW) + pair * 32;          // y handoff

  const int rowbase = blockIdx.x * 32 + pair * 16;

  // ---- build weights in LDS (A_MAT/B_VEC/s are module constants in the ref) ----
  for (int idx = tid; idx < 64 * 64; idx += 128) {
    int r = idx >> 6, c = idx & 63;
    float sr  = sqrtf(2.f * r + 1.f);
    float sc2 = sqrtf(2.f * c + 1.f);
    float sgn = (c <= r) ? 1.f : (((r + c) & 1) ? -1.f : 1.f);
    float a   = -(1.f / DELTAC) * sr * sc2 * sgn;          // A[r][c]
    Et[r * LDSW + c]  = ((r == c) ? 1.f : 0.f) + DTC * a;  // delta + DT*A[n=r][k=c]
    W1t[r * LDSW + c] = W1[c * HIDN + r];
    U1t[r * LDSW + c] = U1[c * HIDN + r];
  }
  for (int idx = lane + (wave & 1) * 32; idx < 16 * LDSW; idx += 64) {
    Hb[idx] = 0.f; Hp[idx] = 0.f;
  }
  if (lane < 16 && isW0) ybuf[lane] = 0.f;
  __syncthreads();

  // ---- role-dependent loop-invariant B fragments: 128 VGPRs ----
  const float* Wsel = isW0 ? Et : W1t;
  v2f wf[4][16];
  #pragma unroll
  for (int nt = 0; nt < 4; ++nt)
    #pragma unroll
    for (int kc = 0; kc < 16; ++kc)
      wf[nt][kc] = *(const v2f*)&Wsel[(nt * 16 + lmod) * LDSW + kc * 4 + 2 * lhalf];

  // ---- per-lane column fragments (index n = nt*16 + lmod) ----
  float b1f[4], w2f[4], dtbf[4], sf[4], ub1f[4], u2f[4];
  #pragma unroll
  for (int nt = 0; nt < 4; ++nt) {
    int n = nt * 16 + lmod;
    float sn = sqrtf(2.f * n + 1.f);
    sf[nt]   = sn;                    // SQRT_CN
    dtbf[nt] = DTC * sn / DELTAC;     // DT * B_VEC
    b1f[nt]  = b1[n];
    w2f[nt]  = W2[n];
    ub1f[nt] = ub1[n];
    u2f[nt]  = U2[n];
  }
  const float b2s = b2[0], ub2s = ub2[0];

  auto loadA = [&](const float* buf, v2f* af) {
    #pragma unroll
    for (int kc = 0; kc < 16; ++kc)
      af[kc] = *(const v2f*)&buf[lmod * LDSW + kc * 4 + 2 * lhalf];
  };
  auto zero8 = [&](v8f* acc) {
    #pragma unroll
    for (int nt = 0; nt < 4; ++nt) {
      v8f zz;
      #pragma unroll
      for (int i = 0; i < 8; ++i) zz[i] = 0.f;
      acc[nt] = zz;
    }
  };
  // GEMM with even/odd K split -> 8 independent WMMA chains (hides XDL latency)
  auto gemm2 = [&](const v2f (*Bf)[16], const v2f* af, v8f* x0, v8f* x1) {
    #pragma unroll
    for (int nt = 0; nt < 4; ++nt)
      #pragma unroll
      for (int kc = 0; kc < 16; kc += 2) {
        x0[nt] = wmma4(af[kc],     Bf[nt][kc],     x0[nt]);
        x1[nt] = wmma4(af[kc + 1], Bf[nt][kc + 1], x1[nt]);
      }
  };
  auto gemm_lds = [&](const float* Bt, const v2f* af, v8f* acc) {
    #pragma unroll
    for (int nt = 0; nt < 4; ++nt)
      #pragma unroll
      for (int kc = 0; kc < 16; ++kc) {
        v2f bf = *(const v2f*)&Bt[(nt * 16 + lmod) * LDSW + kc * 4 + 2 * lhalf];
        acc[nt] = wmma4(af[kc], bf, acc[nt]);
      }
  };
  auto rowdot = [&](const float* buf, const float* cf, float bias, float* op) {
    #pragma unroll
    for (int v = 0; v < 8; ++v) {
      float p = 0.f;
      #pragma unroll
      for (int nt = 0; nt < 4; ++nt)
        p += buf[(v + 8 * lhalf) * LDSW + nt * 16 + lmod] * cf[nt];
      op[v] = halfred(p) + bias;
    }
  };

  // ---- per-role straight-line flow loops (barrier counts match pairwise) ----
  auto w0_flow = [&](int nsteps) {
    for (int s = 0; s < nsteps; ++s) {
      v2f af[16];
      loadA(Hb, af);
      v8f e0[4], e1[4];
      zero8(e0); zero8(e1);
      gemm2(wf, af, e0, e1);           // evolve: Hb @ (I + DT*A^T)
      __syncthreads();                  // (1) ybuf ready; all Hb reads done
      float yr[8];
      #pragma unroll
      for (int v = 0; v < 8; v += 2) {
        v2f yp = *(const v2f*)&ybuf[v + 8 * lhalf];
        yr[v] = yp[0]; yr[v + 1] = yp[1];
      }
      #pragma unroll
      for (int nt = 0; nt < 4; ++nt)
        #pragma unroll
        for (int v = 0; v < 8; ++v)
          Hb[(v + 8 * lhalf) * LDSW + nt * 16 + lmod] =
              e0[nt][v] + e1[nt][v] + dtbf[nt] * yr[v];
      __syncthreads();                  // (2) new Hb visible
    }
  };
  auto w1_flow = [&](int nsteps) {
    for (int s = 0; s < nsteps; ++s) {
      v2f af[16];
      loadA(Hb, af);
      v8f z0[4], z1[4];
      #pragma unroll
      for (int nt = 0; nt < 4; ++nt) {
        v8f a0;
        #pragma unroll
        for (int i = 0; i < 8; ++i) a0[i] = b1f[nt];
        z0[nt] = a0;
      }
      zero8(z1);
      gemm2(wf, af, z0, z1);           // hidden: Hb @ W1 + b1
      #pragma unroll
      for (int v = 0; v < 8; ++v) {
        float p = 0.f;
        #pragma unroll
        for (int nt = 0; nt < 4; ++nt)
          p += fmaxf(z0[nt][v] + z1[nt][v], 0.f) * w2f[nt];
        float y = halfred(p) + b2s;    // y[m], row-broadcast
        if (lmod == 0) ybuf[v + 8 * lhalf] = y;
      }
      __syncthreads();                  // (1)
      __syncthreads();                  // (2)
    }
  };

  const int UNC_OFF = BATCHN * TSTEPS + BATCHN * NCH + TSTEPS; // 73736
  const int H_OFF   = BATCHN * TSTEPS;                         // 8192

  for (int t = 0; t < TSTEPS; ++t) {
    if (isW0) w0_flow(NPRE); else w1_flow(NPRE);

    // save h_pre (pair-split copy); next flow's barrier (1) orders copy vs stores
    for (int idx = lane + (wave & 1) * 32; idx < 16 * LDSW; idx += 64) Hp[idx] = Hb[idx];
    float ypred[8];
    if (isW0) rowdot(Hb, sf, 0.f, ypred);

    if (isW0) w0_flow(NPOST); else w1_flow(NPOST);

    // ---- heads on h_end: w0 reads pred, w1 loads A-frags for unc ----
    float pred[8];
    v2f afu[16];
    if (isW0) rowdot(Hb, sf, 0.f, pred);
    else      loadA(Hb, afu);
    __syncthreads();                   // all Hb reads done before w0 rewrites it

    if (isW0) {
      if (lmod == 0) {
        #pragma unroll
        for (int v = 0; v < 8; ++v)
          out[(rowbase + v + 8 * lhalf) * TSTEPS + t] = pred[v];
      }
      float yobs[8], mrow[8];
      #pragma unroll
      for (int v = 0; v < 8; ++v) {
        int row = rowbase + v + 8 * lhalf;
        yobs[v] = Y[row * TSTEPS + t];
        mrow[v] = mask[row * TSTEPS + t];
      }
      // 10 assimilation steps on Hp (pair-private, no sync needed inside)
      for (int k = 0; k < NUPD; ++k) {
        float frac = 0.1f * (float)k;  // DT*k/CORR
        v2f af[16];
        loadA(Hp, af);
        v8f e0[4], e1[4];
        zero8(e0); zero8(e1);
        gemm2(wf, af, e0, e1);
        #pragma unroll
        for (int nt = 0; nt < 4; ++nt)
          #pragma unroll
          for (int v = 0; v < 8; ++v) {
            float yest = ypred[v] + frac * (yobs[v] - ypred[v]);
            Hp[(v + 8 * lhalf) * LDSW + nt * 16 + lmod] =
                e0[nt][v] + e1[nt][v] + dtbf[nt] * yest;
          }
      }
      // h_new = mask*h_upd + (1-mask)*h_end
      #pragma unroll
      for (int nt = 0; nt < 4; ++nt)
        #pragma unroll
        for (int v = 0; v < 8; ++v) {
          int off = (v + 8 * lhalf) * LDSW + nt * 16 + lmod;
          Hb[off] = Hp[off] * mrow[v] + Hb[off] * (1.f - mrow[v]);
        }
    } else {
      // unc = sigmoid(mlp(h_end, U1, ub1, U2, ub2))
      v8f z[4];
      #pragma unroll
      for (int nt = 0; nt < 4; ++nt) {
        v8f a0;
        #pragma unroll
        for (int i = 0; i < 8; ++i) a0[i] = ub1f[nt];
        z[nt] = a0;
      }
      gemm_lds(U1t, afu, z);
      #pragma unroll
      for (int v = 0; v < 8; ++v) {
        float p = 0.f;
        #pragma unroll
        for (int nt = 0; nt < 4; ++nt) p += fmaxf(z[nt][v], 0.f) * u2f[nt];
        float x = halfred(p) + ub2s;
        float u = 1.f / (1.f + expf(-x));
        if (lmod == 0)
          out[UNC_OFF + (rowbase + v + 8 * lhalf) * TSTEPS + t] = u;
      }
    }
    __syncthreads();                   // merged Hb visible before next outer step
  }

  // final hidden state (w0 writes its pair's rows)
  if (isW0) {
    #pragma unroll
    for (int nt = 0; nt < 4; ++nt)
      #pragma unroll
      for (int v = 0; v < 8; ++v) {
        int row = rowbase + v + 8 * lhalf;
        out[H_OFF + row * NCH + nt * 16 + lmod] =
            Hb[(v + 8 * lhalf) * LDSW + nt * 16 + lmod];
      }
  }

  // times passthrough
  if (blockIdx.x == 0 && tid < TSTEPS)
    out[H_OFF + BATCHN * NCH + tid] = times[tid];
}

extern "C" void kernel_launch(void* const* d_in, const int* in_sizes, int n_in,
                              void* d_out, int out_size, void* d_ws, size_t ws_size,
                              hipStream_t stream) {
  const float* times = (const float*)d_in[0];
  const float* Y     = (const float*)d_in[1];
  const float* mask  = (const float*)d_in[2];
  const float* W1    = (const float*)d_in[3];
  const float* b1    = (const float*)d_in[4];
  const float* W2    = (const float*)d_in[5];
  const float* b2    = (const float*)d_in[6];
  const float* U1    = (const float*)d_in[7];
  const float* ub1   = (const float*)d_in[8];
  const float* U2    = (const float*)d_in[9];
  const float* ub2   = (const float*)d_in[10];

  // LDS: 3 weight matrices + per-pair Hb/Hp + per-pair y handoff
  size_t shmem = (size_t)(3 * 64 * LDSW + 4 * 16 * LDSW + 2 * 32) * sizeof(float); // 69,888 B
  cnode_kernel<<<BATCHN / 32, 128, shmem, stream>>>(
      times, Y, mask, W1, b1, W2, b2, U1, ub1, U2, ub2, (float*)d_out);
}